// VqRouter_37847251812687
// MI455X (gfx1250) — compile-verified
//
#include <hip/hip_runtime.h>

// Problem constants (match reference)
static constexpr int NBT = 2048;   // B*T
static constexpr int ND  = 1024;   // D
static constexpr int NH  = 4;      // hashes
static constexpr int NG  = 4;      // groups
static constexpr int NK  = 128;    // codebook size
static constexpr int NGD = 32;     // group dim
static constexpr int LOGITS_PER_SIDE = NBT * NH * NG * NK;  // 4,194,304
static constexpr int IDX_PER_SIDE    = NBT * NH;            // 8,192

typedef __attribute__((ext_vector_type(2))) float v2f;
typedef __attribute__((ext_vector_type(8))) float v8f;

// One wave (32 lanes) per (side, 16-token tile, h, g).
// Phase 1: y_tile[16 x 32] = tag_tile @ W_slice^T via f32 WMMA over D.
// Phase 2: logits[16 x 128] = 2*(y_tile @ c^T) - ||y||^2 - ||c||^2 via f32 WMMA.
__global__ __launch_bounds__(32)
void vq_gemm_dist_kernel(const float* __restrict__ tag,
                         const float* __restrict__ Wr,
                         const float* __restrict__ Ww,
                         const float* __restrict__ cbr,
                         const float* __restrict__ cbw,
                         float* __restrict__ out)
{
    const int lane = threadIdx.x;        // wave32
    const int li   = lane & 15;
    const int half = lane >> 4;

    const int btt  = blockIdx.x;         // 0..127 token tile
    const int hg   = blockIdx.y;         // 0..15  (h*NG + g)
    const int side = blockIdx.z;         // 0: read side, 1: write side

    const float* W  = side ? Ww : Wr;
    const float* cb = (side ? cbw : cbr) + (size_t)hg * (NK * NGD);
    float* logits   = out + 2 * IDX_PER_SIDE + (size_t)side * LOGITS_PER_SIDE;

    const int bt0   = btt * 16;
    const int wrow0 = hg * NGD;          // = h*128 + g*32 (row of W / column of y)

    // A-fragment lane layout (16x4 f32): lane(li,half) holds row li, K = 2*half, 2*half+1.
    // B-fragment mirrors it.  tag and W are both row-major over D -> identical float2 loads.
    const float* arow  = tag + (size_t)(bt0 + li) * ND + 2 * half;
    const float* brow0 = W + (size_t)(wrow0 + li) * ND + 2 * half;
    const float* brow1 = W + (size_t)(wrow0 + 16 + li) * ND + 2 * half;

    v8f cf0 = {};  // y columns 0..15 of the 32-wide group slice
    v8f cf1 = {};  // y columns 16..31
    #pragma unroll 4
    for (int d = 0; d < ND; d += 4) {
        v2f a  = *(const v2f*)(arow + d);
        v2f b0 = *(const v2f*)(brow0 + d);
        v2f b1 = *(const v2f*)(brow1 + d);
        cf0 = __builtin_amdgcn_wmma_f32_16x16x4_f32(false, a, false, b0, (short)0, cf0, false, false);
        cf1 = __builtin_amdgcn_wmma_f32_16x16x4_f32(false, a, false, b1, (short)0, cf1, false, false);
    }

    // ||y||^2 per token row.  C layout: lane(li,half), vgpr v -> row v+8*half, col li.
    // Reduce squares across the 16 li-lanes (xor bits 0..3 stay within a half).
    float y2[8];
    #pragma unroll
    for (int v = 0; v < 8; ++v) y2[v] = cf0[v] * cf0[v] + cf1[v] * cf1[v];
    #pragma unroll
    for (int m = 1; m < 16; m <<= 1) {
        #pragma unroll
        for (int v = 0; v < 8; ++v) y2[v] += __shfl_xor(y2[v], m, 32);
    }

    // Re-shape y tile C-layout -> A-layout through LDS.
    __shared__ float ytile[16 * NGD];
    #pragma unroll
    for (int v = 0; v < 8; ++v) {
        const int row = v + 8 * half;
        ytile[row * NGD + li]      = cf0[v];
        ytile[row * NGD + 16 + li] = cf1[v];
    }
    __syncthreads();

    v2f af[8];  // A fragments for the 8 depth steps over GD=32
    #pragma unroll
    for (int kk = 0; kk < 8; ++kk)
        af[kk] = *(const v2f*)&ytile[li * NGD + 4 * kk + 2 * half];

    // Distance GEMM: 8 tiles of 16 codes each.
    for (int nt = 0; nt < 8; ++nt) {
        const float* crow = cb + (size_t)(nt * 16 + li) * NGD + 2 * half;
        v8f  dacc = {};
        float c2p = 0.f;
        #pragma unroll
        for (int kk = 0; kk < 8; ++kk) {
            v2f bv = *(const v2f*)(crow + 4 * kk);
            dacc = __builtin_amdgcn_wmma_f32_16x16x4_f32(false, af[kk], false, bv, (short)0, dacc, false, false);
            c2p += bv.x * bv.x + bv.y * bv.y;
        }
        const float c2 = c2p + __shfl_xor(c2p, 16, 32);  // combine the two K-halves

        #pragma unroll
        for (int v = 0; v < 8; ++v) {
            const int row = v + 8 * half;                // token within tile
            const float L = 2.f * dacc[v] - y2[v] - c2;  // -||y - c||^2
            const size_t off = (size_t)(bt0 + row) * (NH * NG * NK) + (size_t)hg * NK + nt * 16 + li;
            logits[off] = L;
        }
    }
}

// One thread per (side, token, h): top-2 per group, combo rule, bucket.
__global__ void vq_scan_kernel(float* __restrict__ out)
{
    const int tid = blockIdx.x * blockDim.x + threadIdx.x;
    if (tid >= 2 * NBT * NH) return;
    const int side = tid >> 13;          // / (NBT*NH)
    const int rem  = tid & 8191;
    const int bt   = rem >> 2;
    const int h    = rem & 3;

    const float* lg = out + 2 * IDX_PER_SIDE + (size_t)side * LOGITS_PER_SIDE
                      + (size_t)bt * (NH * NG * NK) + h * (NG * NK);
    int code[NG];
    for (int g = 0; g < NG; ++g) {
        const float* v = lg + g * NK;
        float m1 = -3.402823466e38f, m2 = -3.402823466e38f;
        int i1 = 0, i2 = 0;
        for (int k = 0; k < NK; ++k) {
            const float x = v[k];
            if (x > m1)      { m2 = m1; i2 = i1; m1 = x; i1 = k; }
            else if (x > m2) { m2 = x;  i2 = k; }
        }
        // combo[g,0] = (g/2)%2 -> groups 0,1 take top-1; groups 2,3 take top-2
        code[g] = (g < 2) ? i1 : i2;
    }
    const long long s = (long long)code[0] + 128LL * code[1]
                      + 16384LL * code[2] + 2097152LL * code[3];
    const int bucket = (int)(s % 65536);
    out[(size_t)side * IDX_PER_SIDE + bt * NH + h] = (float)bucket;
}

extern "C" void kernel_launch(void* const* d_in, const int* in_sizes, int n_in,
                              void* d_out, int out_size, void* d_ws, size_t ws_size,
                              hipStream_t stream) {
    (void)in_sizes; (void)n_in; (void)out_size; (void)d_ws; (void)ws_size;
    const float* tag = (const float*)d_in[0];
    const float* Wr  = (const float*)d_in[1];
    const float* Ww  = (const float*)d_in[2];
    const float* cbr = (const float*)d_in[3];
    const float* cbw = (const float*)d_in[4];
    float* out = (float*)d_out;

    dim3 grid(NBT / 16, NH * NG, 2);     // 128 token tiles x 16 (h,g) x 2 sides
    vq_gemm_dist_kernel<<<grid, 32, 0, stream>>>(tag, Wr, Ww, cbr, cbw, out);

    const int n = 2 * NBT * NH;          // 16384 scan threads
    vq_scan_kernel<<<(n + 255) / 256, 256, 0, stream>>>(out);
}